// SelfAttention_39444979646642
// MI455X (gfx1250) — compile-verified
//
#include <hip/hip_runtime.h>
#include <hip/hip_bf16.h>

// ---------------------------------------------------------------------------
// Self-attention (SAGAN-style, with the reference's exact reshape semantics)
// B=4, C=32, H=120, W=160, NH=8, hd=20, n=C*H=3840
//
// All GEMMs use V_WMMA_F32_16X16X4_F32 (full fp32, wave32, gfx1250).
// ---------------------------------------------------------------------------

typedef float v2f __attribute__((ext_vector_type(2)));
typedef float v8f __attribute__((ext_vector_type(8)));

#define WMMA_F32X4(a, b, c) \
  __builtin_amdgcn_wmma_f32_16x16x4_f32(false, (a), false, (b), (short)0, (c), false, false)

#define NSEQ   3840      // C*H
#define HD     20        // head_dim
#define HDPAD  32        // padded head_dim (two 16-row WMMA tiles)
#define WDIM   160
#define BHF    76800     // per-(b,head) floats in Q/K/V view = HD*NSEQ
#define BF     614400    // per-batch floats = C*H*W

// ---------------------------------------------------------------------------
// Kernel 1: projection  out = x @ Wm^T + bias   (rows=15360, N=160, K=160)
// One wave computes one 16x16 output tile; K-loop = 40 x wmma_f32_16x16x4.
// A[M,K] = x[row, k] ; B[K,N] = Wm[w, k]  (Wm is (out,in) row-major -> pairs
// along k are contiguous for both operands -> float2 loads).
// ---------------------------------------------------------------------------
__global__ __launch_bounds__(32)
void qkv_proj(const float* __restrict__ x, const float* __restrict__ Wm,
              const float* __restrict__ bias, float* __restrict__ out) {
  const int lane = threadIdx.x;
  const int half = lane >> 4;   // K sub-offset selector
  const int col  = lane & 15;   // M row (for A) / N col (for B)
  const int row0 = blockIdx.x * 16;
  const int w0   = blockIdx.y * 16;

  const float* xr = x  + (size_t)(row0 + col) * WDIM;
  const float* wr = Wm + (size_t)(w0   + col) * WDIM;

  v8f acc = {};
  #pragma unroll 4
  for (int kb = 0; kb < WDIM; kb += 4) {
    v2f a = *(const v2f*)(xr + kb + 2 * half);   // A: K=kb+2h, kb+2h+1
    v2f b = *(const v2f*)(wr + kb + 2 * half);   // B: K=kb+2h, kb+2h+1
    acc = WMMA_F32X4(a, b, acc);
  }

  const float bw = bias[w0 + col];
  #pragma unroll
  for (int r = 0; r < 8; ++r) {
    out[(size_t)(row0 + r + 8 * half) * WDIM + w0 + col] = acc[r] + bw;
  }
}

// ---------------------------------------------------------------------------
// Kernel 2: per-query-row softmax stats (online max/sum, never spills E).
// For each (b,head) and 16 query rows: stream 240 key tiles, 5 WMMAs each,
// bounce the D-layout tile through LDS, lanes 0..15 reduce one row each.
// ---------------------------------------------------------------------------
__global__ __launch_bounds__(32)
void softmax_stats(const float* __restrict__ Q, const float* __restrict__ K,
                   float* __restrict__ RM, float* __restrict__ Z) {
  __shared__ float lds[16 * 17];
  const int lane = threadIdx.x;
  const int half = lane >> 4;
  const int col  = lane & 15;
  const int bh   = blockIdx.x;          // 0..31  (b*8+head)
  const int m0   = blockIdx.y * 16;     // query tile base

  const float* q = Q + (size_t)bh * BHF;   // view (HD, NSEQ)
  const float* k = K + (size_t)bh * BHF;

  // A[M=query, K=d] = q[d, m0+M]  (loop-invariant -> preload 5 fragments)
  v2f afrag[5];
  #pragma unroll
  for (int kk = 0; kk < 5; ++kk) {
    const int d = 4 * kk + 2 * half;
    afrag[kk].x = q[(size_t)d * NSEQ + m0 + col];
    afrag[kk].y = q[(size_t)(d + 1) * NSEQ + m0 + col];
  }

  float runmax = -__builtin_inff();
  float runsum = 0.0f;

  for (int j0 = 0; j0 < NSEQ; j0 += 16) {
    v8f e = {};
    #pragma unroll
    for (int kk = 0; kk < 5; ++kk) {
      const int d = 4 * kk + 2 * half;
      v2f b;                                   // B[K=d, N=key] = k[d, j0+N]
      b.x = k[(size_t)d * NSEQ + j0 + col];
      b.y = k[(size_t)(d + 1) * NSEQ + j0 + col];
      e = WMMA_F32X4(afrag[kk], b, e);
    }
    #pragma unroll
    for (int r = 0; r < 8; ++r) lds[(r + 8 * half) * 17 + col] = e[r];
    __syncthreads();
    if (lane < 16) {                           // lane owns query row m0+lane
      const float* row = lds + lane * 17;
      float tmax = row[0];
      #pragma unroll
      for (int c = 1; c < 16; ++c) tmax = fmaxf(tmax, row[c]);
      const float nm = fmaxf(runmax, tmax);
      float s = 0.0f;
      #pragma unroll
      for (int c = 0; c < 16; ++c) s += __expf(row[c] - nm);
      runsum = runsum * __expf(runmax - nm) + s;
      runmax = nm;
    }
    __syncthreads();
  }
  if (lane < 16) {
    RM[(size_t)bh * NSEQ + m0 + lane] = runmax;
    Z [(size_t)bh * NSEQ + m0 + lane] = runsum;
  }
}

// ---------------------------------------------------------------------------
// Kernel 3: vp[bh, dpad, m] = (dpad < 20) ? V[bh, dpad, m] / Z[bh, m] : 0
// (fold 1/Z into V; pad d to 32 so pass 2 tiles cleanly)
// ---------------------------------------------------------------------------
__global__ __launch_bounds__(256)
void vp_build(const float* __restrict__ V, const float* __restrict__ Z,
              float* __restrict__ VP) {
  const int idx = blockIdx.x * 256 + threadIdx.x;   // over 32*32*3840
  if (idx >= 32 * HDPAD * NSEQ) return;
  const int m  = idx % NSEQ;
  const int t  = idx / NSEQ;
  const int dp = t % HDPAD;
  const int bh = t / HDPAD;
  float v = 0.0f;
  if (dp < HD)
    v = V[(size_t)bh * BHF + (size_t)dp * NSEQ + m] / Z[(size_t)bh * NSEQ + m];
  VP[idx] = v;
}

// ---------------------------------------------------------------------------
// Kernel 4: out[d, n] = sum_m vp[d, m] * exp(E[m, n] - RM[m])
// Per (bh, 16-col n-tile): stream m in 16-chunks; 5 WMMAs rebuild E tile,
// exp+rowmax applied in D-layout, tile bounced via LDS into B-layout,
// 2x4 WMMAs accumulate the (padded 32-row) vp x P product.
// Final store applies the reference's transpose(0,2,1,3).reshape mapping.
// ---------------------------------------------------------------------------
__global__ __launch_bounds__(32)
void attn_out(const float* __restrict__ Q, const float* __restrict__ K,
              const float* __restrict__ VP, const float* __restrict__ RM,
              float* __restrict__ out) {
  __shared__ float lds[16 * 17];
  const int lane = threadIdx.x;
  const int half = lane >> 4;
  const int col  = lane & 15;
  const int bh   = blockIdx.x;
  const int n0   = blockIdx.y * 16;
  const int b    = bh >> 3;
  const int head = bh & 7;

  const float* q  = Q  + (size_t)bh * BHF;
  const float* k  = K  + (size_t)bh * BHF;
  const float* vp = VP + (size_t)bh * (HDPAD * NSEQ);
  const float* rm = RM + (size_t)bh * NSEQ;

  // B[K=d, N=n] = k[d, n0+N]  (loop-invariant -> preload)
  v2f bfrag[5];
  #pragma unroll
  for (int kk = 0; kk < 5; ++kk) {
    const int d = 4 * kk + 2 * half;
    bfrag[kk].x = k[(size_t)d * NSEQ + n0 + col];
    bfrag[kk].y = k[(size_t)(d + 1) * NSEQ + n0 + col];
  }

  v8f acc0 = {};   // d = 0..15
  v8f acc1 = {};   // d = 16..31 (rows 20..31 multiply zero-padded vp)

  for (int m0 = 0; m0 < NSEQ; m0 += 16) {
    // E[m, n] tile
    v8f e = {};
    #pragma unroll
    for (int kk = 0; kk < 5; ++kk) {
      const int d = 4 * kk + 2 * half;
      v2f a;                                  // A[M=m, K=d] = q[d, m0+M]
      a.x = q[(size_t)d * NSEQ + m0 + col];
      a.y = q[(size_t)(d + 1) * NSEQ + m0 + col];
      e = WMMA_F32X4(a, bfrag[kk], e);
    }
    // rowmax for rows m0+8*half .. +7 (uniform broadcast loads per half)
    const float4* rmp = (const float4*)(rm + m0 + 8 * half);
    const float4 r01 = rmp[0];
    const float4 r23 = rmp[1];
    const float rmv[8] = {r01.x, r01.y, r01.z, r01.w, r23.x, r23.y, r23.z, r23.w};
    #pragma unroll
    for (int r = 0; r < 8; ++r)
      lds[(r + 8 * half) * 17 + col] = __expf(e[r] - rmv[r]);
    __syncthreads();
    // accumulate vp (A, rows=d) x P (B, rows=m-chunk)
    #pragma unroll
    for (int kk = 0; kk < 4; ++kk) {
      const int mk = 4 * kk + 2 * half;
      v2f bp;                                 // B[K=mk, N=n] from LDS tile
      bp.x = lds[mk * 17 + col];
      bp.y = lds[(mk + 1) * 17 + col];
      v2f a0 = *(const v2f*)(vp + (size_t)col * NSEQ + m0 + mk);         // d=col
      v2f a1 = *(const v2f*)(vp + (size_t)(16 + col) * NSEQ + m0 + mk);  // d=16+col
      acc0 = WMMA_F32X4(a0, bp, acc0);
      acc1 = WMMA_F32X4(a1, bp, acc1);
    }
    __syncthreads();
  }

  // out_flat[b, d*NH*n + head*n + (n0+col)]  (transpose(0,2,1,3).reshape)
  float* ob = out + (size_t)b * BF + (size_t)head * NSEQ + n0 + col;
  #pragma unroll
  for (int r = 0; r < 8; ++r) {
    const int d = r + 8 * half;               // 0..15
    ob[(size_t)d * (8 * NSEQ)] = acc0[r];
  }
  if (half == 0) {
    #pragma unroll
    for (int r = 0; r < 4; ++r) {
      const int d = 16 + r;                   // 16..19 valid only
      ob[(size_t)d * (8 * NSEQ)] = acc1[r];
    }
  }
}

// ---------------------------------------------------------------------------
// Host launch
// ---------------------------------------------------------------------------
extern "C" void kernel_launch(void* const* d_in, const int* in_sizes, int n_in,
                              void* d_out, int out_size, void* d_ws, size_t ws_size,
                              hipStream_t stream) {
  const float* x  = (const float*)d_in[0];
  const float* Wq = (const float*)d_in[1];
  const float* bq = (const float*)d_in[2];
  const float* Wk = (const float*)d_in[3];
  const float* bk = (const float*)d_in[4];
  const float* Wv = (const float*)d_in[5];
  const float* bv = (const float*)d_in[6];
  float* out = (float*)d_out;

  float* ws = (float*)d_ws;
  float* Q  = ws;                 // 4*614400
  float* Km = ws + 2457600;       // 4*614400
  float* V  = ws + 4915200;       // 4*614400
  float* Z  = ws + 7372800;       // 32*3840
  float* RM = ws + 7495680;       // 32*3840
  float* VP = ws + 7618560;       // 32*32*3840 (padded head dim)

  const dim3 projGrid(15360 / 16, WDIM / 16);   // (960, 10)
  qkv_proj<<<projGrid, 32, 0, stream>>>(x, Wq, bq, Q);
  qkv_proj<<<projGrid, 32, 0, stream>>>(x, Wk, bk, Km);
  qkv_proj<<<projGrid, 32, 0, stream>>>(x, Wv, bv, V);

  softmax_stats<<<dim3(32, NSEQ / 16), 32, 0, stream>>>(Q, Km, RM, Z);

  vp_build<<<(32 * HDPAD * NSEQ + 255) / 256, 256, 0, stream>>>(V, Z, VP);

  attn_out<<<dim3(32, NSEQ / 16), 32, 0, stream>>>(Q, Km, VP, RM, out);
}